// dense_unet_19524921328214
// MI455X (gfx1250) — compile-verified
//
#include <hip/hip_runtime.h>

typedef float v2f __attribute__((ext_vector_type(2)));
typedef float v8f __attribute__((ext_vector_type(8)));

#define NEGS 0.2f
#define EPSB 1e-5f

__device__ __forceinline__ float lrelu_f(float v) { return v > 0.f ? v : v * NEGS; }

// ---------------------------------------------------------------------------
// dine: out[n, colOff+c] = sum_j gather7(src, no)[n, j] * W[j, c] + b[c]
// A tile (16 rows x 7*CC) staged in LDS per channel-chunk CC (<=64);
// 4 waves x 16 cols via V_WMMA_F32_16X16X4_F32. Cout <= 64.
// ---------------------------------------------------------------------------
__global__ void __launch_bounds__(128)
dine_wmma(const float* __restrict__ src, const int* __restrict__ no,
          const float* __restrict__ W, const float* __restrict__ bias,
          float* __restrict__ out, int N, int Cin, int Cout,
          int outStride, int colOff)
{
    __shared__ float lds[16 * 452];           // 16 x (7*64 + 4) floats = 28.9 KB
    const int CC  = (Cin < 64) ? Cin : 64;    // channel chunk
    const int KC  = 7 * CC;                   // K per chunk (multiple of 4)
    const int lst = KC + 4;                   // padded LDS row stride
    const int nChunks = Cin / CC;
    const int row0 = blockIdx.x * 16;
    const int tid  = threadIdx.x;

    const int wave = tid >> 5;
    const int lane = tid & 31;
    const int half = lane >> 4;
    const int lr   = lane & 15;
    const int col  = wave * 16 + lr;
    const int wcol = (col < Cout) ? col : (Cout - 1);   // clamp: no branches

    v8f acc = {0.f, 0.f, 0.f, 0.f, 0.f, 0.f, 0.f, 0.f};

    for (int chunk = 0; chunk < nChunks; ++chunk) {
        const int c0 = chunk * CC;
        __syncthreads();
        // cooperative gather of A chunk (float4 granularity)
        const int total4 = (16 * KC) >> 2;
        for (int idx = tid; idx < total4; idx += 128) {
            int lin = idx << 2;
            int r   = lin / KC;
            int jl  = lin - r * KC;
            float4 v = make_float4(0.f, 0.f, 0.f, 0.f);
            int grow = row0 + r;
            if (grow < N) {
                int k  = jl / CC;
                int cc = jl - k * CC;         // multiple of 4
                int node = no[grow * 7 + k];
                v = *(const float4*)(src + (size_t)node * Cin + c0 + cc);
            }
            *(float4*)(lds + r * lst + jl) = v;
        }
        __syncthreads();

        const float* lrow = lds + lr * lst + 2 * half;
        for (int k = 0; k < 7; ++k) {
            const float* lA = lrow + k * CC;
            const float* Wl = W + (size_t)(k * Cin + c0 + 2 * half) * Cout + wcol;
            const size_t wstep = (size_t)4 * Cout;
            #pragma unroll 4
            for (int cc = 0; cc < CC; cc += 4) {
                v2f a; a[0] = lA[cc]; a[1] = lA[cc + 1];
                v2f bb; bb[0] = Wl[0]; bb[1] = Wl[Cout];
                acc = __builtin_amdgcn_wmma_f32_16x16x4_f32(
                    false, a, false, bb, (short)0, acc, false, false);
                Wl += wstep;
            }
        }
    }

    if (col < Cout) {
        float bv = bias[col];
        for (int v = 0; v < 8; ++v) {
            int row = row0 + v + 8 * half;    // D: M = v + 8*half, N = lr
            if (row < N)
                out[(size_t)row * outStride + colOff + col] = acc[v] + bv;
        }
    }
}

// ---------------------------------------------------------------------------
// upconv matmul: y(raw,448) = x(raw,64) @ W(64,448) + b
// grid.y = 7 column blocks of 64; 4 waves x 16 cols each.
// ---------------------------------------------------------------------------
__global__ void __launch_bounds__(128)
upmat_wmma(const float* __restrict__ x, const float* __restrict__ W,
           const float* __restrict__ bias, float* __restrict__ y, int raw)
{
    __shared__ float lds[16 * 68];
    const int row0 = blockIdx.x * 16;
    const int tid  = threadIdx.x;
    for (int idx = tid; idx < 16 * 64; idx += 128) {
        int r = idx >> 6, c = idx & 63;
        int grow = row0 + r;
        lds[r * 68 + c] = (grow < raw) ? x[(size_t)grow * 64 + c] : 0.f;
    }
    __syncthreads();

    const int wave = tid >> 5, lane = tid & 31;
    const int half = lane >> 4, lr = lane & 15;
    const int col  = blockIdx.y * 64 + wave * 16 + lr;   // < 448

    v8f acc = {0.f, 0.f, 0.f, 0.f, 0.f, 0.f, 0.f, 0.f};
    const float* lA = lds + lr * 68 + 2 * half;
    const float* Wl = W + (size_t)(2 * half) * 448 + col;
    #pragma unroll 4
    for (int kk = 0; kk < 64; kk += 4) {
        v2f a; a[0] = lA[kk]; a[1] = lA[kk + 1];
        v2f bb; bb[0] = Wl[0]; bb[1] = Wl[448];
        acc = __builtin_amdgcn_wmma_f32_16x16x4_f32(
            false, a, false, bb, (short)0, acc, false, false);
        Wl += (size_t)4 * 448;
    }
    float bv = bias[col];
    for (int v = 0; v < 8; ++v) {
        int row = row0 + v + 8 * half;
        if (row < raw) y[(size_t)row * 448 + col] = acc[v] + bv;
    }
}

// ---------------------------------------------------------------------------
// batchnorm helpers
// ---------------------------------------------------------------------------
__global__ void zero_f32(float* p, int n)
{
    int i = blockIdx.x * blockDim.x + threadIdx.x;
    if (i < n) p[i] = 0.f;
}

__global__ void __launch_bounds__(256)
bn_partial(const float* __restrict__ in, int N, int C, int inStride,
           float* __restrict__ gsum, float* __restrict__ gsq)
{
    __shared__ float s[512];                 // sums[C], sumsq[C], C <= 256
    const int tid = threadIdx.x;
    for (int c = tid; c < 2 * C; c += 256) s[c] = 0.f;
    __syncthreads();
    const int base  = blockIdx.x * 64;
    const int total = 64 * C;
    for (int idx = tid; idx < total; idx += 256) {
        int r = base + idx / C;
        int c = idx % C;
        if (r < N) {
            float v = in[(size_t)r * inStride + c];
            atomicAdd(&s[c], v);
            atomicAdd(&s[C + c], v * v);
        }
    }
    __syncthreads();
    for (int c = tid; c < C; c += 256) {
        atomicAdd(&gsum[c], s[c]);
        atomicAdd(&gsq[c],  s[C + c]);
    }
}

__global__ void __launch_bounds__(256)
bn_apply_lrelu(const float* __restrict__ in, float* __restrict__ out,
               int N, int C, int inStride, int outStride,
               const float* __restrict__ gsum, const float* __restrict__ gsq,
               const float* __restrict__ g, const float* __restrict__ be)
{
    int idx = blockIdx.x * 256 + threadIdx.x;
    if (idx >= N * C) return;
    int r = idx / C, c = idx % C;
    float invN = 1.f / (float)N;
    float mu   = gsum[c] * invN;
    float var  = gsq[c] * invN - mu * mu;
    float rstd = rsqrtf(var + EPSB);
    float v = (in[(size_t)r * inStride + c] - mu) * rstd * g[c] + be[c];
    out[(size_t)r * outStride + c] = lrelu_f(v);
}

// ---------------------------------------------------------------------------
// 7-neighborhood mean pool (C = 64)
// ---------------------------------------------------------------------------
__global__ void __launch_bounds__(256)
pool7(const float* __restrict__ x, const int* __restrict__ no,
      float* __restrict__ out, int num, int outStride)
{
    int idx = blockIdx.x * 256 + threadIdx.x;
    if (idx >= num * 64) return;
    int i = idx >> 6, c = idx & 63;
    float s = 0.f;
    for (int k = 0; k < 7; ++k) s += x[(size_t)no[i * 7 + k] * 64 + c];
    out[(size_t)i * outStride + c] = s * (1.f / 7.f);
}

// ---------------------------------------------------------------------------
// upconv combine: out = lrelu(skip + concat(y[top], pairmean(y[down])))
// y viewed as (raw*7, 64) row-major (same flat memory as (raw, 448)).
// ---------------------------------------------------------------------------
__global__ void __launch_bounds__(256)
upcomb(const float* __restrict__ y, const int* __restrict__ top,
       const int* __restrict__ down, const float* __restrict__ skip,
       float* __restrict__ out, int raw, int newN, int outStride)
{
    int idx = blockIdx.x * 256 + threadIdx.x;
    if (idx >= newN * 64) return;
    int i = idx >> 6, c = idx & 63;
    float v;
    if (i < raw) {
        v = y[(size_t)top[i] * 64 + c];
    } else {
        int ii = i - raw;
        int d  = (c < 32) ? down[2 * ii] : down[2 * ii + 1];
        int cc = (2 * c) & 63;
        v = 0.5f * (y[(size_t)d * 64 + cc] + y[(size_t)d * 64 + cc + 1]);
    }
    v = lrelu_f(skip[(size_t)i * 64 + c] + v);
    out[(size_t)i * outStride + c] = v;
}

// ---------------------------------------------------------------------------
// host orchestration
// ---------------------------------------------------------------------------
extern "C" void kernel_launch(void* const* d_in, const int* in_sizes, int n_in,
                              void* d_out, int out_size, void* d_ws, size_t ws_size,
                              hipStream_t stream)
{
    (void)in_sizes; (void)n_in; (void)out_size; (void)ws_size;
    const int N0 = 40962, N1 = 10242, N2 = 2562, N3 = 642;

    const float* x  = (const float*)d_in[0];
    const int* n40  = (const int*)d_in[1];
    const int* n10  = (const int*)d_in[2];
    const int* n25  = (const int*)d_in[3];
    const int* n6   = (const int*)d_in[4];
    const int* t25  = (const int*)d_in[5];
    const int* d25  = (const int*)d_in[6];
    const int* t10  = (const int*)d_in[7];
    const int* d10  = (const int*)d_in[8];
    const int* t40  = (const int*)d_in[9];
    const int* d40  = (const int*)d_in[10];

    int pi = 11;
    auto P = [&]() { return (const float*)d_in[pi++]; };
    // down1: c1{W,b}, bn1{g,be}, c2{W,b}, bn2{g,be}
    const float *d1c1W = P(), *d1c1b = P(), *d1bn1g = P(), *d1bn1be = P();
    const float *d1c2W = P(), *d1c2b = P(), *d1bn2g = P(), *d1bn2be = P();
    // dense1..dense5: 4 x { bn{g,be}, conv{W,b} }
    const float *dg[5][4], *dbe[5][4], *dW[5][4], *db[5][4];
    for (int bq = 0; bq < 5; ++bq)
        for (int i = 0; i < 4; ++i) {
            dg[bq][i] = P(); dbe[bq][i] = P(); dW[bq][i] = P(); db[bq][i] = P();
        }
    const float *u1W = P(), *u1b = P(), *u2W = P(), *u2b = P(), *u3W = P(), *u3b = P();
    const float *oc1W = P(), *oc1b = P(), *obng = P(), *obnbe = P(), *oc2W = P(), *oc2b = P();

    // workspace carving (floats)
    float* wsf = (float*)d_ws;
    size_t off = 0;
    auto alloc = [&](size_t n) { float* p = wsf + off; off += n; return p; };
    float* stat = alloc(512);                    // gsum[256] + gsq[256]
    float* X1 = alloc((size_t)N0 * 64);
    float* X2 = alloc((size_t)N1 * 64);
    float* X3 = alloc((size_t)N2 * 64);
    float* X4 = alloc((size_t)N3 * 64);
    float* F  = alloc((size_t)N1 * 256);         // dense feats (stride 256)
    float* T  = alloc((size_t)N1 * 256);         // bn temp, also >= N0*64
    float* U  = alloc((size_t)N1 * 448);         // upconv y
    float* H  = alloc((size_t)N1 * 64);          // dense4/5 outputs
    float* G  = alloc((size_t)N0 * 64);          // generic N0x64

    auto run_dine = [&](const float* src, const int* no, const float* W,
                        const float* bb, float* out, int N, int Cin, int Cout,
                        int outStride, int colOff) {
        int grid = (N + 15) / 16;
        dine_wmma<<<grid, 128, 0, stream>>>(src, no, W, bb, out, N, Cin, Cout,
                                            outStride, colOff);
    };
    auto run_bn = [&](const float* in, float* out, int N, int C,
                      int inStride, int outStride, const float* g, const float* be) {
        zero_f32<<<2, 256, 0, stream>>>(stat, 512);
        int nb = (N + 63) / 64;
        bn_partial<<<nb, 256, 0, stream>>>(in, N, C, inStride, stat, stat + 256);
        int tot = N * C;
        bn_apply_lrelu<<<(tot + 255) / 256, 256, 0, stream>>>(
            in, out, N, C, inStride, outStride, stat, stat + 256, g, be);
    };
    auto run_pool = [&](const float* src, const int* no, float* out, int num, int ostr) {
        int tot = num * 64;
        pool7<<<(tot + 255) / 256, 256, 0, stream>>>(src, no, out, num, ostr);
    };
    auto run_dense = [&](const int* no, int N, int bq, float* outBuf) {
        for (int i = 0; i < 4; ++i) {
            int Ci = 64 * (i + 1);
            run_bn(F, T, N, Ci, 256, Ci, dg[bq][i], dbe[bq][i]);
            if (i < 3) run_dine(T, no, dW[bq][i], db[bq][i], F, N, Ci, 64, 256, Ci);
            else       run_dine(T, no, dW[bq][i], db[bq][i], outBuf, N, Ci, 64, 64, 0);
        }
    };
    auto run_up = [&](const float* hin, const float* W, const float* bb,
                      const int* top, const int* down, const float* skip,
                      float* out, int raw, int newN, int outStride) {
        dim3 g((raw + 15) / 16, 7);
        upmat_wmma<<<g, 128, 0, stream>>>(hin, W, bb, U, raw);
        int tot = newN * 64;
        upcomb<<<(tot + 255) / 256, 256, 0, stream>>>(U, top, down, skip, out,
                                                      raw, newN, outStride);
    };

    // ---- down1 ----
    run_dine(x, n40, d1c1W, d1c1b, G, N0, 4, 64, 64, 0);
    run_bn(G, G, N0, 64, 64, 64, d1bn1g, d1bn1be);          // h
    run_dine(G, n40, d1c2W, d1c2b, T, N0, 64, 64, 64, 0);
    run_bn(T, X1, N0, 64, 64, 64, d1bn2g, d1bn2be);         // x1

    // ---- encoder ----
    run_pool(X1, n40, F, N1, 256);
    run_dense(n10, N1, 0, X2);                               // dense1 -> x2
    run_pool(X2, n10, F, N2, 256);
    run_dense(n25, N2, 1, X3);                               // dense2 -> x3
    run_pool(X3, n25, F, N3, 256);
    run_dense(n6,  N3, 2, X4);                               // dense3 -> x4

    // ---- decoder ----
    run_up(X4, u1W, u1b, t25, d25, X3, F, N3, N2, 256);      // h -> feats(2562)
    run_dense(n25, N2, 3, H);                                // dense4
    run_up(H, u2W, u2b, t10, d10, X2, F, N2, N1, 256);       // h -> feats(10242)
    run_dense(n10, N1, 4, H);                                // dense5
    run_up(H, u3W, u3b, t40, d40, X1, G, N1, N0, 64);        // h (40962x64)

    // ---- outc ----
    run_dine(G, n40, oc1W, oc1b, T, N0, 64, 64, 64, 0);
    run_bn(T, T, N0, 64, 64, 64, obng, obnbe);
    run_dine(T, n40, oc2W, oc2b, (float*)d_out, N0, 64, 40, 40, 0);
}